// QuantumHybridModel_77558519431282
// MI455X (gfx1250) — compile-verified
//
#include <hip/hip_runtime.h>

typedef __attribute__((ext_vector_type(2))) float v2f;
typedef __attribute__((ext_vector_type(8))) float v8f;

#define NQ 4
#define NL 2
#define DIM 20
#define HID 64
#define TOK_PER_WAVE 32
#define WAVES 4
#define TOK_PER_BLOCK 128
#define HSTRIDE 68   // 64 + 4 pad: conflict-friendly LDS row stride

__global__ __launch_bounds__(TOK_PER_BLOCK) void qhyb_fused(
    const float* __restrict__ x,        // [B*S, 20]
    const float* __restrict__ W_embed,  // [64, 20]
    const float* __restrict__ b_embed,  // [64]
    const float* __restrict__ W_c2q,    // [4, 64]
    const float* __restrict__ b_c2q,    // [4]
    const float* __restrict__ q_weights,// [2, 4, 3]
    const float* __restrict__ W_q2c,    // [64, 4]
    const float* __restrict__ b_q2c,    // [64]
    const float* __restrict__ W_out,    // [3, 64]
    const float* __restrict__ b_out,    // [3]
    float* __restrict__ out)            // [B*S, 3]
{
    __shared__ float h_lds[TOK_PER_BLOCK * HSTRIDE];
    __shared__ float wc2q_lds[NQ * HID];
    __shared__ float bemb_lds[HID];
    __shared__ float u_lds[NL][NQ][8];   // 2x2 complex: 00r,00i,01r,01i,10r,10i,11r,11i
    __shared__ float M_lds[3][NQ];       // W_out @ W_q2c
    __shared__ float c_lds[3];           // W_out @ b_q2c + b_out

    const int tid = threadIdx.x;

    // ---------------- Phase 0: per-block constant precompute ----------------
    for (int i = tid; i < NQ * HID; i += TOK_PER_BLOCK) wc2q_lds[i] = W_c2q[i];
    if (tid < HID) bemb_lds[tid] = b_embed[tid];
    if (tid < NL * NQ) {
        int layer = tid >> 2, q = tid & 3;
        float phi = q_weights[(layer * NQ + q) * 3 + 0];
        float th  = q_weights[(layer * NQ + q) * 3 + 1];
        float om  = q_weights[(layer * NQ + q) * 3 + 2];
        float ct = cosf(0.5f * th), st = sinf(0.5f * th);
        float cp = cosf(0.5f * (phi + om)), sp = sinf(0.5f * (phi + om));
        float cm = cosf(0.5f * (phi - om)), sm = sinf(0.5f * (phi - om));
        float* U = &u_lds[layer][q][0];
        U[0] =  cp * ct; U[1] = -sp * ct;   // u00 = ep*c
        U[2] = -cm * st; U[3] = -sm * st;   // u01 = -conj(em)*s
        U[4] =  cm * st; U[5] = -sm * st;   // u10 = em*s
        U[6] =  cp * ct; U[7] =  sp * ct;   // u11 = conj(ep)*c
    }
    if (tid >= 8 && tid < 8 + 12) {
        int o = (tid - 8) >> 2, q = (tid - 8) & 3;
        float s = 0.0f;
        for (int h = 0; h < HID; ++h) s += W_out[o * HID + h] * W_q2c[h * NQ + q];
        M_lds[o][q] = s;
    }
    if (tid >= 20 && tid < 23) {
        int o = tid - 20;
        float s = b_out[o];
        for (int h = 0; h < HID; ++h) s += W_out[o * HID + h] * b_q2c[h];
        c_lds[o] = s;
    }
    __syncthreads();

    // ---------------- Phase 1: embed GEMM via f32 WMMA ----------------
    const int lane = tid & 31;
    const int wave = tid >> 5;
    const int lo = lane & 15, hi = lane >> 4;
    const long blockTok0 = (long)blockIdx.x * TOK_PER_BLOCK;
    const int  waveRow0  = wave * TOK_PER_WAVE;
    const long waveTok0  = blockTok0 + waveRow0;

    // Preload B fragments (W_embed): B[k][n] = W_embed[n][k0+k]
    // lane L holds B[2*hi + {0,1}][lo] for each (kstep, ntile)
    v2f bfrag[5][4];
#pragma unroll
    for (int ks = 0; ks < 5; ++ks)
#pragma unroll
        for (int n = 0; n < 4; ++n) {
            int hcol = n * 16 + lo;
            int d0 = ks * 4 + 2 * hi;
            bfrag[ks][n].x = W_embed[hcol * DIM + d0];
            bfrag[ks][n].y = W_embed[hcol * DIM + d0 + 1];
        }

    v8f acc[2][4];
#pragma unroll
    for (int mt = 0; mt < 2; ++mt)
#pragma unroll
        for (int n = 0; n < 4; ++n)
            acc[mt][n] = (v8f){0.f, 0.f, 0.f, 0.f, 0.f, 0.f, 0.f, 0.f};

#pragma unroll
    for (int ks = 0; ks < 5; ++ks) {
#pragma unroll
        for (int mt = 0; mt < 2; ++mt) {
            long t = waveTok0 + mt * 16 + lo;   // A row = token
            int d0 = ks * 4 + 2 * hi;
            v2f afrag;
            afrag.x = x[t * DIM + d0];
            afrag.y = x[t * DIM + d0 + 1];
#pragma unroll
            for (int n = 0; n < 4; ++n)
                acc[mt][n] = __builtin_amdgcn_wmma_f32_16x16x4_f32(
                    false, afrag, false, bfrag[ks][n],
                    (short)0, acc[mt][n], false, false);
        }
    }

    // Spill C fragments with bias + ReLU: C elem (M = r + 8*hi, N = lo) per n-tile
#pragma unroll
    for (int mt = 0; mt < 2; ++mt)
#pragma unroll
        for (int n = 0; n < 4; ++n) {
            int hc = n * 16 + lo;
            float bb = bemb_lds[hc];
#pragma unroll
            for (int r = 0; r < 8; ++r) {
                int row = waveRow0 + mt * 16 + r + 8 * hi;
                h_lds[row * HSTRIDE + hc] = fmaxf(acc[mt][n][r] + bb, 0.0f);
            }
        }
    __syncthreads();

    // ---------------- Phase 2: per-token tail (1 lane = 1 token) ----------------
    const int myRow = waveRow0 + lane;
    const long tok = blockTok0 + myRow;
    const float* hrow = &h_lds[myRow * HSTRIDE];

    // q_in = tanh(W_c2q @ h + b_c2q)
    float cs[NQ], sn[NQ];
#pragma unroll
    for (int q = 0; q < NQ; ++q) {
        float s = b_c2q[q];
#pragma unroll 16
        for (int k = 0; k < HID; ++k) s += wc2q_lds[q * HID + k] * hrow[k];
        float a = 0.5f * tanhf(s);
        cs[q] = cosf(a);
        sn[q] = sinf(a);
    }

    // Initial product state: idx bit3=q0, bit2=q1, bit1=q2, bit0=q3
    float ar[16], ai[16];
#pragma unroll
    for (int i = 0; i < 16; ++i) {
        float v = ((i >> 3) & 1 ? sn[0] : cs[0]) *
                  ((i >> 2) & 1 ? sn[1] : cs[1]) *
                  ((i >> 1) & 1 ? sn[2] : cs[2]) *
                  ((i      ) & 1 ? sn[3] : cs[3]);
        ar[i] = v; ai[i] = 0.0f;
    }

#pragma unroll
    for (int layer = 0; layer < NL; ++layer) {
#pragma unroll
        for (int q = 0; q < NQ; ++q) {
            const float* U = &u_lds[layer][q][0];
            float u00r = U[0], u00i = U[1], u01r = U[2], u01i = U[3];
            float u10r = U[4], u10i = U[5], u11r = U[6], u11i = U[7];
            const int p = 3 - q;
#pragma unroll
            for (int i = 0; i < 16; ++i) {
                if (i & (1 << p)) continue;
                int j = i | (1 << p);
                float s0r = ar[i], s0i = ai[i], s1r = ar[j], s1i = ai[j];
                ar[i] = u00r * s0r - u00i * s0i + u01r * s1r - u01i * s1i;
                ai[i] = u00r * s0i + u00i * s0r + u01r * s1i + u01i * s1r;
                ar[j] = u10r * s0r - u10i * s0i + u11r * s1r - u11i * s1i;
                ai[j] = u10r * s0i + u10i * s0r + u11r * s1i + u11i * s1r;
            }
        }
        // CNOT chain: control q (bit 3-q), target q+1 (bit 2-q)
#pragma unroll
        for (int q = 0; q < NQ - 1; ++q) {
            const int cbit = 1 << (3 - q), tbit = 1 << (2 - q);
#pragma unroll
            for (int i = 0; i < 16; ++i) {
                if ((i & cbit) && !(i & tbit)) {
                    int j = i | tbit;
                    float tr = ar[i], ti = ai[i];
                    ar[i] = ar[j]; ai[i] = ai[j];
                    ar[j] = tr;    ai[j] = ti;
                }
            }
        }
    }

    // <Z_q> expectations
    float z[NQ] = {0.f, 0.f, 0.f, 0.f};
#pragma unroll
    for (int i = 0; i < 16; ++i) {
        float p = ar[i] * ar[i] + ai[i] * ai[i];
#pragma unroll
        for (int q = 0; q < NQ; ++q)
            z[q] += (i & (1 << (3 - q))) ? -p : p;
    }

    // coords = M @ z + c   (fused W_q2c, b_q2c, W_out, b_out)
#pragma unroll
    for (int o = 0; o < 3; ++o) {
        float v = c_lds[o];
#pragma unroll
        for (int q = 0; q < NQ; ++q) v += M_lds[o][q] * z[q];
        out[tok * 3 + o] = v;
    }
}

extern "C" void kernel_launch(void* const* d_in, const int* in_sizes, int n_in,
                              void* d_out, int out_size, void* d_ws, size_t ws_size,
                              hipStream_t stream) {
    const float* x        = (const float*)d_in[0];
    const float* W_embed  = (const float*)d_in[1];
    const float* b_embed  = (const float*)d_in[2];
    const float* W_c2q    = (const float*)d_in[3];
    const float* b_c2q    = (const float*)d_in[4];
    const float* q_w      = (const float*)d_in[5];
    const float* W_q2c    = (const float*)d_in[6];
    const float* b_q2c    = (const float*)d_in[7];
    const float* W_out    = (const float*)d_in[8];
    const float* b_out    = (const float*)d_in[9];
    float* out = (float*)d_out;

    const int tokens = in_sizes[0] / DIM;           // B*S = 262144
    const int blocks = tokens / TOK_PER_BLOCK;      // 2048
    qhyb_fused<<<blocks, TOK_PER_BLOCK, 0, stream>>>(
        x, W_embed, b_embed, W_c2q, b_c2q, q_w, W_q2c, b_q2c, W_out, b_out, out);
}